// BPLoss_19963007991831
// MI455X (gfx1250) — compile-verified
//
#include <hip/hip_runtime.h>
#include <math.h>
#include <stdint.h>

// BPLoss: N=100000 detections, C=1024 classes, M=128 GTs.
// out = -sum(z*log(rowmax(masked cs))) - sum(r*log(...)) + exp(-sum(z*dist))
// HBM-bound: 409.6 MB stream of class_scores -> ~17.6us floor at 23.3 TB/s.
// Strategy: one wave32 per row; async global->LDS b128 staging, double-buffered,
// ASYNCcnt-paced; deterministic two-stage reduction (no float atomics).

#define BLK 128          // 4 waves of 32
#define WPB 4            // waves per block
#define MAXBLOCKS 1024
#define CCOLS 1024       // class_scores row width (floats)

__device__ __forceinline__ void issue_row_async(const float* rowPtr, uint32_t ldsAddr, int lane)
{
    // per-lane global address; imm offset advances BOTH lds and global addrs
    uint64_t ga = (uint64_t)(uintptr_t)rowPtr + (uint64_t)(lane * 16);
#define ASYNC_LD(OFF) \
    asm volatile("global_load_async_to_lds_b128 %0, %1, off offset:" OFF \
                 :: "v"(ldsAddr), "v"(ga) : "memory")
    ASYNC_LD("0");    ASYNC_LD("512");  ASYNC_LD("1024"); ASYNC_LD("1536");
    ASYNC_LD("2048"); ASYNC_LD("2560"); ASYNC_LD("3072"); ASYNC_LD("3584");
#undef ASYNC_LD
}

__global__ void __launch_bounds__(BLK) bploss_main(
    const float* __restrict__ cs,    // [N, 1024]
    const float* __restrict__ xywh,  // [N, 4]
    const float* __restrict__ z,     // [N]
    const float* __restrict__ r,     // [N]
    const int*   __restrict__ nidx,  // [N]
    const int*   __restrict__ gl,    // [M]
    const float* __restrict__ gxy,   // [M, 4]
    float* __restrict__ partials,    // [3 * nblocks]
    int N, int nblocks)
{
    __shared__ float smem[WPB * 2048];   // 2 x 4KB double buffer per wave = 32KB
    __shared__ float wZ[WPB], wR[WPB], wD[WPB];

    const int lane       = threadIdx.x & 31;
    const int wave       = threadIdx.x >> 5;
    const int wavesTotal = gridDim.x * WPB;
    const int waveGlobal = blockIdx.x * WPB + wave;

    // LDS byte address for this lane (flat-ptr low 32 bits == LDS offset)
    uint32_t ldsBase = (uint32_t)(uintptr_t)(&smem[wave * 2048]) + (uint32_t)(lane * 16);

    float accZ = 0.f, accR = 0.f, accD = 0.f;

    int  row  = waveGlobal;
    bool have = row < N;
    if (have) issue_row_async(cs + (size_t)row * CCOLS, ldsBase, lane);
    int buf = 0;

    while (have) {
        const int  nrow  = row + wavesTotal;
        const bool hnext = nrow < N;
        if (hnext)
            issue_row_async(cs + (size_t)nrow * CCOLS, ldsBase + (uint32_t)((buf ^ 1) * 4096), lane);

        // per-row scalar work overlaps the async wait
        const int   gi    = nidx[row];
        const int   label = gl[gi];
        const float zv    = z[row];
        const float rv    = r[row];
        float dist = 0.f;
        if (lane == 0) {
            float dx = xywh[4 * row + 0] - gxy[4 * gi + 0];
            float dy = xywh[4 * row + 1] - gxy[4 * gi + 1];
            float dz = xywh[4 * row + 2] - gxy[4 * gi + 2];
            float dw = xywh[4 * row + 3] - gxy[4 * gi + 3];
            dist = dx * dx + dy * dy + dz * dz + dw * dw;
        }

        // in-order completion: <=8 outstanding means current buffer landed
        if (hnext) asm volatile("s_wait_asynccnt 0x8" ::: "memory");
        else       asm volatile("s_wait_asynccnt 0x0" ::: "memory");

        const float4* ldsRow = (const float4*)&smem[wave * 2048 + buf * 1024];
        float m = -__builtin_inff();
#pragma unroll
        for (int s = 0; s < 8; ++s) {
            float4 v = ldsRow[s * 32 + lane];
            const int c0 = s * 128 + lane * 4;
            v.x = (c0 + 0 == label) ? 0.f : v.x;
            v.y = (c0 + 1 == label) ? 0.f : v.y;
            v.z = (c0 + 2 == label) ? 0.f : v.z;
            v.w = (c0 + 3 == label) ? 0.f : v.w;
            m = fmaxf(m, fmaxf(fmaxf(v.x, v.y), fmaxf(v.z, v.w)));
        }
#pragma unroll
        for (int off = 16; off > 0; off >>= 1)
            m = fmaxf(m, __shfl_xor(m, off, 32));

        if (lane == 0) {
            const float lm = logf(m);
            accZ += zv * lm;
            accR += rv * lm;
            accD += zv * dist;
        }

        row = nrow; have = hnext; buf ^= 1;
    }

    if (lane == 0) { wZ[wave] = accZ; wR[wave] = accR; wD[wave] = accD; }
    __syncthreads();
    if (threadIdx.x == 0) {
        float sZ = 0.f, sR = 0.f, sD = 0.f;
        for (int w = 0; w < WPB; ++w) { sZ += wZ[w]; sR += wR[w]; sD += wD[w]; }
        partials[blockIdx.x]               = sZ;
        partials[nblocks + blockIdx.x]     = sR;
        partials[2 * nblocks + blockIdx.x] = sD;
    }
}

__global__ void __launch_bounds__(256) bploss_finish(const float* __restrict__ partials,
                                                     float* __restrict__ out, int nb)
{
    __shared__ float sZ[256], sR[256], sD[256];
    const int t = threadIdx.x;
    float aZ = 0.f, aR = 0.f, aD = 0.f;
    for (int i = t; i < nb; i += 256) {
        aZ += partials[i];
        aR += partials[nb + i];
        aD += partials[2 * nb + i];
    }
    sZ[t] = aZ; sR[t] = aR; sD[t] = aD;
    __syncthreads();
    for (int off = 128; off > 0; off >>= 1) {
        if (t < off) { sZ[t] += sZ[t + off]; sR[t] += sR[t + off]; sD[t] += sD[t + off]; }
        __syncthreads();
    }
    if (t == 0)
        out[0] = -sZ[0] - sR[0] + expf(-sD[0]);  // tpc + fpc + tps
}

extern "C" void kernel_launch(void* const* d_in, const int* in_sizes, int n_in,
                              void* d_out, int out_size, void* d_ws, size_t ws_size,
                              hipStream_t stream)
{
    const float* cs   = (const float*)d_in[0];
    const float* xywh = (const float*)d_in[1];
    const float* z    = (const float*)d_in[2];
    const float* r    = (const float*)d_in[3];
    const int*   nidx = (const int*)d_in[4];
    const int*   gl   = (const int*)d_in[5];
    const float* gxy  = (const float*)d_in[6];

    const int N = in_sizes[2];  // z is [N]

    int nblocks = MAXBLOCKS;
    if (ws_size < (size_t)3 * nblocks * sizeof(float)) {
        nblocks = (int)(ws_size / (3 * sizeof(float)));
        if (nblocks < 1) nblocks = 1;
    }
    float* partials = (float*)d_ws;

    bploss_main<<<nblocks, BLK, 0, stream>>>(cs, xywh, z, r, nidx, gl, gxy,
                                             partials, N, nblocks);
    bploss_finish<<<1, 256, 0, stream>>>(partials, (float*)d_out, nblocks);
}